// HEPT_75196287418964
// MI455X (gfx1250) — compile-verified
//
#include <hip/hip_runtime.h>

typedef __attribute__((ext_vector_type(2))) float v2f;
typedef __attribute__((ext_vector_type(8))) float v8f;

#define BH     16          // B*H
#define SEQ    2048
#define DIM    64
#define TILES  (SEQ / 16)  // 128 tiles per dimension

// Pass 1: qs[bh*S+i] = -0.5*||q_i||^2 ; ks[bh*S+j] = -0.5*||k_j||^2
__global__ __launch_bounds__(256) void hept_norms(const float* __restrict__ q,
                                                  const float* __restrict__ k,
                                                  float* __restrict__ qs,
                                                  float* __restrict__ ks) {
    int t = blockIdx.x * 256 + threadIdx.x;      // 0 .. 2*BH*SEQ-1
    int which = t >> 15;                          // BH*SEQ = 32768
    int idx = t & (BH * SEQ - 1);
    const float4* row = (const float4*)((which ? k : q) + (size_t)idx * DIM);
    float acc = 0.f;
#pragma unroll
    for (int d = 0; d < DIM / 4; ++d) {
        float4 v = row[d];
        acc += v.x * v.x + v.y * v.y + v.z * v.z + v.w * v.w;
    }
    (which ? ks : qs)[idx] = -0.5f * acc;
}

// Pass 2: one wave per 16x16 output tile; 16 x V_WMMA_F32_16X16X4_F32 over K=64,
// then rank-1 corrections + clamp + exp, store fp32.
__global__ __launch_bounds__(256) void hept_gemm_exp(const float* __restrict__ q,
                                                     const float* __restrict__ kk,
                                                     const float* __restrict__ qs,
                                                     const float* __restrict__ ks,
                                                     float* __restrict__ out) {
    const int lane = threadIdx.x & 31;
    const int wave = blockIdx.x * (256 >> 5) + (threadIdx.x >> 5);

    const int tj = wave & (TILES - 1);
    const int ti = (wave >> 7) & (TILES - 1);
    const int bh = wave >> 14;

    const int lo = lane & 15;   // row within A tile / col within B,C,D tiles
    const int hi = lane >> 4;   // selects K-pair (operands) / row-half (C,D)

    // A lane data: Q[ti*16+lo][kb + 2*hi .. +1]   (contiguous float2)
    // B lane data: K[tj*16+lo][kb + 2*hi .. +1]   (same pattern -> no transpose)
    const float* Qrow = q  + ((size_t)bh * SEQ + (size_t)ti * 16 + lo) * DIM + 2 * hi;
    const float* Krow = kk + ((size_t)bh * SEQ + (size_t)tj * 16 + lo) * DIM + 2 * hi;

    v8f c = {};
#pragma unroll
    for (int kb = 0; kb < DIM; kb += 4) {
        v2f a = *(const v2f*)(Qrow + kb);
        v2f b = *(const v2f*)(Krow + kb);
        // D = A(16x4,f32) x B(4x16,f32) + C ; 8-arg form like other f32-acc WMMAs
        c = __builtin_amdgcn_wmma_f32_16x16x4_f32(
                /*neg_a=*/false, a, /*neg_b=*/false, b,
                /*c_mod=*/(short)0, c, /*reuse_a=*/false, /*reuse_b=*/false);
    }

    // Epilogue: C/D layout -> VGPR v holds row (v + 8*hi), col lo.
    const float  ksv  = ks[(size_t)bh * SEQ + (size_t)tj * 16 + lo];
    const float* qsp  = qs + (size_t)bh * SEQ + (size_t)ti * 16 + 8 * hi;
    float* obase = out + ((size_t)bh * SEQ + (size_t)ti * 16 + 8 * hi) * SEQ
                       + (size_t)tj * 16 + lo;
#pragma unroll
    for (int v = 0; v < 8; ++v) {
        float logit = c[v] + qsp[v] + ksv;
        logit = fminf(logit, 0.0f);
        obase[(size_t)v * SEQ] = __expf(logit);
    }
}

extern "C" void kernel_launch(void* const* d_in, const int* in_sizes, int n_in,
                              void* d_out, int out_size, void* d_ws, size_t ws_size,
                              hipStream_t stream) {
    const float* q  = (const float*)d_in[0];
    const float* k  = (const float*)d_in[1];
    float* out = (float*)d_out;

    float* qs = (float*)d_ws;            // BH*SEQ floats
    float* ks = qs + (size_t)BH * SEQ;   // BH*SEQ floats (total 256 KB of d_ws)

    // Pass 1: 2*BH*SEQ = 65536 threads
    hept_norms<<<(2 * BH * SEQ) / 256, 256, 0, stream>>>(q, k, qs, ks);

    // Pass 2: BH * TILES * TILES waves = 262144 waves, 8 waves (256 thr) per block
    const int nblocks = (BH * TILES * TILES) / 8;  // 32768
    hept_gemm_exp<<<nblocks, 256, 0, stream>>>(q, k, qs, ks, out);
}